// StandardSAGE_43980465111675
// MI455X (gfx1250) — compile-verified
//
#include <hip/hip_runtime.h>
#include <hip/hip_bf16.h>

typedef __attribute__((ext_vector_type(2))) float v2f;
typedef __attribute__((ext_vector_type(8))) float v8f;

#define N_NODES 50000
#define N_EDGES 600000
#define DIM     128

// ---------------------------------------------------------------- utilities
__global__ void sage_zero_kernel(float* __restrict__ p, int n) {
    int i = blockIdx.x * blockDim.x + threadIdx.x;
    if (i < n) p[i] = 0.0f;
}

// in-degree counts (float, so the mean path stays in VALU)
__global__ void sage_count_kernel(const int* __restrict__ ei, float* __restrict__ cnt, int E) {
    int e = blockIdx.x * blockDim.x + threadIdx.x;
    if (e < E) atomicAdd(&cnt[ei[E + e]], 1.0f);
}

// edge-parallel scatter: 32 threads per edge, each moves a float4 chunk.
// Gather of x[src] is a fully-coalesced 512B row; scatter is 4 float atomics
// (global_atomic_add_f32) into the dst row.
__global__ void sage_aggregate_kernel(const float* __restrict__ xin,
                                      const int* __restrict__ ei,
                                      float* __restrict__ agg, int E) {
    int idx = blockIdx.x * blockDim.x + threadIdx.x;
    if (idx >= E * (DIM / 4)) return;
    int e = idx >> 5;           // DIM/4 == 32 chunks per edge
    int c = idx & 31;
    int src = ei[e];
    int dst = ei[E + e];
    const float4 v = ((const float4*)(xin + (size_t)src * DIM))[c];
    float* d = agg + (size_t)dst * DIM + (size_t)c * 4;
    atomicAdd(d + 0, v.x);
    atomicAdd(d + 1, v.y);
    atomicAdd(d + 2, v.z);
    atomicAdd(d + 3, v.w);
}

// ---------------------------------------------------------------- WMMA GEMM
// Block = 8 waves; blockIdx.x selects a 16-node tile, wave w owns output
// column tile w. Each wave computes a 16(M) x 16(N) tile of
//   out = act( (agg/cnt) @ Wl^T + b + xin @ Wr^T )
// K = 128, via 2 x 32 V_WMMA_F32_16X16X4_F32 into one f32 accumulator.
//
// A layout (16x4 f32, ISA 7.12.2): lane&15 = row m; K offset = 2*(lane>>4)+vgpr.
// B layout mirrors A with lane&15 = column n (B = W^T, so b[v] = W[col][k]).
// C/D layout: vgpr r holds (m = r + 8*(lane>>4), n = lane&15).
//
// FULL=true  : all 16 columns valid -> branch-free, B loads are b64.
// FULL=false : columns masked multiplicatively (no EXEC divergence around WMMA).
template <bool FULL, bool RELU>
__global__ __launch_bounds__(256)
void sage_wmma_gemm_kernel(const float* __restrict__ agg,
                           const float* __restrict__ cnt,
                           const float* __restrict__ xin,
                           const float* __restrict__ Wl,
                           const float* __restrict__ Wr,
                           const float* __restrict__ bias,
                           float* __restrict__ out,
                           int ncols, int ldo) {
    const int lane = threadIdx.x & 31;
    const int wave = threadIdx.x >> 5;
    const int q    = lane & 15;
    const int half = lane >> 4;

    const int row = blockIdx.x * 16 + q;                 // node index (A loads)
    const int col = wave * 16 + q;                       // output column (B loads)
    const bool colOk = FULL || (col < ncols);
    const float mask = colOk ? 1.0f : 0.0f;
    const int  colc  = FULL ? col : (colOk ? col : 0);   // clamped, in-bounds

    const float invc = 1.0f / fmaxf(cnt[row], 1.0f);     // mean folded into A
    const float* aRow  = agg + (size_t)row * DIM;
    const float* xRow  = xin + (size_t)row * DIM;
    const float* wlRow = Wl + (size_t)colc * DIM;
    const float* wrRow = Wr + (size_t)colc * DIM;

    v8f acc = {};
    // (agg/cnt) @ Wl^T
    #pragma unroll
    for (int kk = 0; kk < DIM; kk += 4) {
        const int ka = kk + half * 2;
        v2f a, b;
        a.x = aRow[ka]     * invc;
        a.y = aRow[ka + 1] * invc;
        b.x = wlRow[ka];
        b.y = wlRow[ka + 1];
        if (!FULL) { b.x *= mask; b.y *= mask; }
        acc = __builtin_amdgcn_wmma_f32_16x16x4_f32(false, a, false, b,
                                                    (short)0, acc, false, false);
    }
    // xin @ Wr^T accumulated into the same tile
    #pragma unroll
    for (int kk = 0; kk < DIM; kk += 4) {
        const int ka = kk + half * 2;
        v2f a, b;
        a.x = xRow[ka];
        a.y = xRow[ka + 1];
        b.x = wrRow[ka];
        b.y = wrRow[ka + 1];
        if (!FULL) { b.x *= mask; b.y *= mask; }
        acc = __builtin_amdgcn_wmma_f32_16x16x4_f32(false, a, false, b,
                                                    (short)0, acc, false, false);
    }

    const float bv = bias[colc] * mask;
    #pragma unroll
    for (int r = 0; r < 8; ++r) {
        float v = acc[r] + bv;
        if (RELU) v = fmaxf(v, 0.0f);
        if (colOk)                                        // store guard only
            out[(size_t)(blockIdx.x * 16 + r + half * 8) * ldo + col] = v;
    }
}

// ---------------------------------------------------------------- launch
extern "C" void kernel_launch(void* const* d_in, const int* in_sizes, int n_in,
                              void* d_out, int out_size, void* d_ws, size_t ws_size,
                              hipStream_t stream) {
    (void)in_sizes; (void)n_in; (void)out_size; (void)ws_size;

    const float* x   = (const float*)d_in[0];
    const int*   ei  = (const int*)d_in[1];     // edge_index [2, E] (int32)
    const float* Wl0 = (const float*)d_in[2];
    const float* Wr0 = (const float*)d_in[3];
    const float* b0  = (const float*)d_in[4];
    const float* Wl1 = (const float*)d_in[5];
    const float* Wr1 = (const float*)d_in[6];
    const float* b1  = (const float*)d_in[7];
    const float* Wl2 = (const float*)d_in[8];
    const float* Wr2 = (const float*)d_in[9];
    const float* b2  = (const float*)d_in[10];
    float* outp = (float*)d_out;

    float* ws  = (float*)d_ws;
    float* agg = ws;                                  // N*DIM
    float* h1  = ws + (size_t)N_NODES * DIM;          // N*DIM
    float* h2  = ws + (size_t)2 * N_NODES * DIM;      // N*DIM
    float* cnt = ws + (size_t)3 * N_NODES * DIM;      // N

    const int feat_elems = N_NODES * DIM;             // 6,400,000
    const int agg_thr    = N_EDGES * (DIM / 4);       // 19,200,000
    const int node_tiles = N_NODES / 16;              // 3125 (exact)

    // in-degree counts (graph fixed across layers)
    sage_zero_kernel<<<(N_NODES + 255) / 256, 256, 0, stream>>>(cnt, N_NODES);
    sage_count_kernel<<<(N_EDGES + 255) / 256, 256, 0, stream>>>(ei, cnt, N_EDGES);

    // layer 0: x -> h1 (relu)
    sage_zero_kernel<<<(feat_elems + 255) / 256, 256, 0, stream>>>(agg, feat_elems);
    sage_aggregate_kernel<<<(agg_thr + 255) / 256, 256, 0, stream>>>(x, ei, agg, N_EDGES);
    sage_wmma_gemm_kernel<true, true><<<node_tiles, 256, 0, stream>>>(
        agg, cnt, x, Wl0, Wr0, b0, h1, DIM, DIM);

    // layer 1: h1 -> h2 (relu)
    sage_zero_kernel<<<(feat_elems + 255) / 256, 256, 0, stream>>>(agg, feat_elems);
    sage_aggregate_kernel<<<(agg_thr + 255) / 256, 256, 0, stream>>>(h1, ei, agg, N_EDGES);
    sage_wmma_gemm_kernel<true, true><<<node_tiles, 256, 0, stream>>>(
        agg, cnt, h1, Wl1, Wr1, b1, h2, DIM, DIM);

    // layer 2: h2 -> out (OUT_DIM = 2, no relu); single wave per node tile
    sage_zero_kernel<<<(feat_elems + 255) / 256, 256, 0, stream>>>(agg, feat_elems);
    sage_aggregate_kernel<<<(agg_thr + 255) / 256, 256, 0, stream>>>(h2, ei, agg, N_EDGES);
    sage_wmma_gemm_kernel<false, false><<<node_tiles, 32, 0, stream>>>(
        agg, cnt, h2, Wl2, Wr2, b2, outp, 2, 2);
}